// StaticGraphGNN_16475494547669
// MI455X (gfx1250) — compile-verified
//
#include <hip/hip_runtime.h>
#include <math.h>

// ---------------------------------------------------------------------------
// StaticGraphGNN (3-layer GAT) for MI455X / gfx1250, wave32.
//  - GEMMs via V_WMMA_F32_16X16X4_F32 (fp32-exact matrix path)
//  - Edge softmax + aggregation via CSR (counting sort by dst) with
//    one wave32 per destination node; no float atomics.
// ---------------------------------------------------------------------------

typedef __attribute__((ext_vector_type(2))) float v2f;
typedef __attribute__((ext_vector_type(4))) float v4f;
typedef __attribute__((ext_vector_type(8))) float v8f;

#define N_NODES   50000
#define N_EDGES   800000
#define EN        (N_EDGES + N_NODES)   // edges + self loops = 850000
#define NEG_SLOPE 0.2f
#define LN_EPS    1e-5f

// ---------------------------------------------------------------------------
// CSR construction: histogram -> exclusive scan -> scatter (srcs sorted by dst)
// ---------------------------------------------------------------------------
__global__ __launch_bounds__(256)
void count_kernel(const int* __restrict__ ei, int* __restrict__ counts) {
  int e = blockIdx.x * blockDim.x + threadIdx.x;
  if (e >= EN) return;
  int dst = (e < N_EDGES) ? ei[N_EDGES + e] : (e - N_EDGES);
  atomicAdd(&counts[dst], 1);
}

__global__ __launch_bounds__(1024)
void scan_kernel(const int* __restrict__ counts, int* __restrict__ row_off) {
  __shared__ int lds[1024];
  const int t  = threadIdx.x;
  const int CH = (N_NODES + 1023) / 1024;  // 49
  const int base = t * CH;
  int s = 0;
  for (int i = 0; i < CH; i++) {
    int idx = base + i;
    if (idx < N_NODES) s += counts[idx];
  }
  lds[t] = s;
  __syncthreads();
  for (int off = 1; off < 1024; off <<= 1) {
    int v = (t >= off) ? lds[t - off] : 0;
    __syncthreads();
    lds[t] += v;
    __syncthreads();
  }
  int run = (t == 0) ? 0 : lds[t - 1];
  for (int i = 0; i < CH; i++) {
    int idx = base + i;
    if (idx < N_NODES) { row_off[idx] = run; run += counts[idx]; }
  }
  if (t == 1023) row_off[N_NODES] = lds[1023];
}

__global__ __launch_bounds__(256)
void scatter_kernel(const int* __restrict__ ei, const int* __restrict__ row_off,
                    int* __restrict__ cursor, int* __restrict__ edge_srcs) {
  int e = blockIdx.x * blockDim.x + threadIdx.x;
  if (e >= EN) return;
  int src, dst;
  if (e < N_EDGES) { src = ei[e]; dst = ei[N_EDGES + e]; }
  else             { src = dst = e - N_EDGES; }
  int pos = row_off[dst] + atomicAdd(&cursor[dst], 1);
  edge_srcs[pos] = src;
}

// ---------------------------------------------------------------------------
// fp32 GEMM: C[M,Nc] = A[M,K] @ B[K,Nc] with V_WMMA_F32_16X16X4_F32.
// One wave32 per 16x16 output tile, 8 waves per block.
// A fragment (16x4 f32): lane m=lane&15, k-half kph=(lane>>4)*2 -> {A[m][k+kph],A[m][k+kph+1]}
// B fragment (4x16 f32): symmetric -> {B[k+kph][n], B[k+kph+1][n]}, n=lane&15
// D (16x16 f32): VGPR j holds M = j + 8*(lane>>4), N = lane&15
// ---------------------------------------------------------------------------
__global__ __launch_bounds__(256)
void gemm_wmma_f32(const float* __restrict__ A, const float* __restrict__ B,
                   float* __restrict__ C, int K, int Nc, int tiles_n, int total_tiles) {
  const int wave = threadIdx.x >> 5;
  const int lane = threadIdx.x & 31;
  const int tile = blockIdx.x * (blockDim.x >> 5) + wave;
  if (tile >= total_tiles) return;  // uniform per wave: EXEC stays all-ones
  const int tm  = tile / tiles_n;
  const int tn  = tile - tm * tiles_n;
  const int m16 = lane & 15;
  const int kph = (lane >> 4) * 2;

  const float* __restrict__ Arow = A + (size_t)(tm * 16 + m16) * K;
  const float* __restrict__ Bcol = B + (tn * 16 + m16);

  v8f acc = {};
  for (int k = 0; k < K; k += 4) {
    v2f a = *(const v2f*)(Arow + k + kph);
    v2f b;
    b.x = Bcol[(size_t)(k + kph)     * Nc];
    b.y = Bcol[(size_t)(k + kph + 1) * Nc];
    acc = __builtin_amdgcn_wmma_f32_16x16x4_f32(
        /*neg_a=*/false, a, /*neg_b=*/false, b,
        /*c_mod=*/(short)0, acc, /*reuse_a=*/false, /*reuse_b=*/false);
  }

  const int ncol  = tn * 16 + m16;
  const int mbase = tm * 16 + (lane >> 4) * 8;
#pragma unroll
  for (int j = 0; j < 8; j++)
    C[(size_t)(mbase + j) * Nc + ncol] = acc[j];
}

// ---------------------------------------------------------------------------
// Per-node attention logits: als[n,h] = sum_c h[n,h,c]*a_s[h,c] (likewise ald).
// One wave per node; a_s/a_d flat layout matches h's channel layout.
// ---------------------------------------------------------------------------
template <int HC, int H>
__global__ __launch_bounds__(256)
void logits_kernel(const float* __restrict__ hfeat, const float* __restrict__ as,
                   const float* __restrict__ ad, float* __restrict__ als,
                   float* __restrict__ ald) {
  const int lane = threadIdx.x & 31;
  const int node = (blockIdx.x * blockDim.x + threadIdx.x) >> 5;
  if (node >= N_NODES) return;
  constexpr int C = HC / H;
  float ps[H], pd[H];
#pragma unroll
  for (int i = 0; i < H; i++) { ps[i] = 0.f; pd[i] = 0.f; }
  const float* hp = hfeat + (size_t)node * HC;
#pragma unroll
  for (int j = 0; j < HC / 32; j++) {
    const int c  = lane + 32 * j;
    const int hh = (32 * j) / C;  // compile-time: 32 | C
    float v = hp[c];
    ps[hh] += v * as[c];
    pd[hh] += v * ad[c];
  }
#pragma unroll
  for (int i = 0; i < H; i++) {
    for (int off = 16; off > 0; off >>= 1) {
      ps[i] += __shfl_xor(ps[i], off, 32);
      pd[i] += __shfl_xor(pd[i], off, 32);
    }
  }
  if (lane == 0) {
#pragma unroll
    for (int i = 0; i < H; i++) {
      als[node * H + i] = ps[i];
      ald[node * H + i] = pd[i];
    }
  }
}

// ---------------------------------------------------------------------------
// GAT aggregation, one wave32 per destination node:
//   pass1: segment max of leaky_relu(als[src]+ald[dst]) per head
//   pass2: softmax denominator
//   pass3: out[n,c] = sum_e exp(e)*h[src,c] / denom + bias[c]
// CPL = channels per lane (HC/32). Coalesced b128 gathers of h[src].
// ---------------------------------------------------------------------------
template <int HC, int H, int CPL>
__global__ __launch_bounds__(256)
void gat_aggregate(const float* __restrict__ hfeat, const float* __restrict__ als,
                   const float* __restrict__ ald, const int* __restrict__ row_off,
                   const int* __restrict__ edge_srcs, const float* __restrict__ bias,
                   float* __restrict__ out) {
  const int lane = threadIdx.x & 31;
  const int node = (blockIdx.x * blockDim.x + threadIdx.x) >> 5;
  if (node >= N_NODES) return;
  const int s = row_off[node];
  const int t = row_off[node + 1];

  float aldn[H];
#pragma unroll
  for (int i = 0; i < H; i++) aldn[i] = ald[node * H + i];

  // pass 1: per-head segment max
  float mx[H];
#pragma unroll
  for (int i = 0; i < H; i++) mx[i] = -INFINITY;
  for (int i = s + lane; i < t; i += 32) {
    const int src = edge_srcs[i];
#pragma unroll
    for (int hh = 0; hh < H; hh++) {
      float e = als[src * H + hh] + aldn[hh];
      e = (e > 0.f) ? e : NEG_SLOPE * e;
      mx[hh] = fmaxf(mx[hh], e);
    }
  }
#pragma unroll
  for (int i = 0; i < H; i++) {
    for (int off = 16; off > 0; off >>= 1)
      mx[i] = fmaxf(mx[i], __shfl_xor(mx[i], off, 32));
    if (mx[i] == -INFINITY) mx[i] = 0.f;  // empty-segment guard (ref semantics)
  }

  // pass 2: softmax denominator
  float den[H];
#pragma unroll
  for (int i = 0; i < H; i++) den[i] = 0.f;
  for (int i = s + lane; i < t; i += 32) {
    const int src = edge_srcs[i];
#pragma unroll
    for (int hh = 0; hh < H; hh++) {
      float e = als[src * H + hh] + aldn[hh];
      e = (e > 0.f) ? e : NEG_SLOPE * e;
      den[hh] += __expf(e - mx[hh]);
    }
  }
#pragma unroll
  for (int i = 0; i < H; i++)
    for (int off = 16; off > 0; off >>= 1)
      den[i] += __shfl_xor(den[i], off, 32);

  // pass 3: weighted feature accumulation (all lanes walk all edges)
  const int   c0   = lane * CPL;
  const int   hl   = c0 / (HC / H);  // constant per lane
  const float mloc = mx[hl];
  const float aldl = aldn[hl];
  float acc[CPL];
#pragma unroll
  for (int j = 0; j < CPL; j++) acc[j] = 0.f;

  for (int i = s; i < t; i++) {
    const int src = edge_srcs[i];
    float e = als[src * H + hl] + aldl;
    e = (e > 0.f) ? e : NEG_SLOPE * e;
    const float w = __expf(e - mloc);
    const float* hp = hfeat + (size_t)src * HC + c0;
    if constexpr (CPL >= 4) {
#pragma unroll
      for (int q = 0; q < CPL / 4; q++) {
        v4f u = ((const v4f*)hp)[q];
        acc[4 * q + 0] += w * u.x;
        acc[4 * q + 1] += w * u.y;
        acc[4 * q + 2] += w * u.z;
        acc[4 * q + 3] += w * u.w;
      }
    } else {
      v2f u = *(const v2f*)hp;
      acc[0] += w * u.x;
      acc[1] += w * u.y;
    }
  }

  const float inv = 1.0f / (den[hl] + 1e-16f);
#pragma unroll
  for (int j = 0; j < CPL; j++)
    out[(size_t)node * HC + c0 + j] = acc[j] * inv + bias[c0 + j];
}

// ---------------------------------------------------------------------------
// Fused LayerNorm(256) + ReLU, in-place; one wave per node.
// ---------------------------------------------------------------------------
__global__ __launch_bounds__(256)
void ln_relu_kernel(float* __restrict__ x, const float* __restrict__ g,
                    const float* __restrict__ b) {
  const int lane = threadIdx.x & 31;
  const int node = (blockIdx.x * blockDim.x + threadIdx.x) >> 5;
  if (node >= N_NODES) return;
  float* xp = x + (size_t)node * 256 + lane * 8;
  v4f u = ((const v4f*)xp)[0];
  v4f w = ((const v4f*)xp)[1];
  float vals[8] = {u.x, u.y, u.z, u.w, w.x, w.y, w.z, w.w};
  float s = 0.f;
#pragma unroll
  for (int j = 0; j < 8; j++) s += vals[j];
  for (int off = 16; off > 0; off >>= 1) s += __shfl_xor(s, off, 32);
  const float mu = s * (1.0f / 256.0f);
  float q = 0.f;
#pragma unroll
  for (int j = 0; j < 8; j++) { float d = vals[j] - mu; q += d * d; }
  for (int off = 16; off > 0; off >>= 1) q += __shfl_xor(q, off, 32);
  const float rstd = rsqrtf(q * (1.0f / 256.0f) + LN_EPS);
  const int c0 = lane * 8;
#pragma unroll
  for (int j = 0; j < 8; j++) {
    float y = (vals[j] - mu) * rstd * g[c0 + j] + b[c0 + j];
    xp[j] = fmaxf(y, 0.f);
  }
}

// ---------------------------------------------------------------------------
// Host-side launch
// ---------------------------------------------------------------------------
extern "C" void kernel_launch(void* const* d_in, const int* in_sizes, int n_in,
                              void* d_out, int out_size, void* d_ws, size_t ws_size,
                              hipStream_t stream) {
  (void)in_sizes; (void)n_in; (void)out_size; (void)ws_size;

  const float* x    = (const float*)d_in[0];
  const int*   ei   = (const int*)d_in[1];   // [2,E] row-major int32
  const float* W1   = (const float*)d_in[2];
  const float* a1s  = (const float*)d_in[3];
  const float* a1d  = (const float*)d_in[4];
  const float* b1   = (const float*)d_in[5];
  const float* ln1g = (const float*)d_in[6];
  const float* ln1b = (const float*)d_in[7];
  const float* W2   = (const float*)d_in[8];
  const float* a2s  = (const float*)d_in[9];
  const float* a2d  = (const float*)d_in[10];
  const float* b2   = (const float*)d_in[11];
  const float* ln2g = (const float*)d_in[12];
  const float* ln2b = (const float*)d_in[13];
  const float* W3   = (const float*)d_in[14];
  const float* a3s  = (const float*)d_in[15];
  const float* a3d  = (const float*)d_in[16];
  const float* b3   = (const float*)d_in[17];
  float* out = (float*)d_out;

  // workspace carving (256B aligned)
  char* base = (char*)d_ws;
  size_t off = 0;
  auto carve = [&](size_t bytes) -> void* {
    void* p = base + off;
    off += (bytes + 255) & ~(size_t)255;
    return p;
  };
  float* h        = (float*)carve((size_t)N_NODES * 256 * 4);
  float* bufA     = (float*)carve((size_t)N_NODES * 256 * 4);
  float* bufB     = (float*)carve((size_t)N_NODES * 256 * 4);
  float* als      = (float*)carve((size_t)N_NODES * 4 * 4);
  float* ald      = (float*)carve((size_t)N_NODES * 4 * 4);
  int*   row_off  = (int*)carve((size_t)(N_NODES + 1) * 4);
  int*   counts   = (int*)carve((size_t)N_NODES * 4);
  int*   edge_srcs = (int*)carve((size_t)EN * 4);

  const int NB_E = (EN + 255) / 256;         // edge-parallel blocks
  const int NB_N = N_NODES / (256 / 32);     // one wave per node: 6250 blocks

  // --- CSR build (deterministic work each call) ---
  hipMemsetAsync(counts, 0, (size_t)N_NODES * 4, stream);
  count_kernel<<<NB_E, 256, 0, stream>>>(ei, counts);
  scan_kernel<<<1, 1024, 0, stream>>>(counts, row_off);
  hipMemsetAsync(counts, 0, (size_t)N_NODES * 4, stream);  // reuse as cursor
  scatter_kernel<<<NB_E, 256, 0, stream>>>(ei, row_off, counts, edge_srcs);

  // --- Layer 1: [N,128]@[128,256] -> GAT -> LN+ReLU ---
  gemm_wmma_f32<<<(N_NODES / 16) * 16 / 8, 256, 0, stream>>>(x, W1, h, 128, 256, 16,
                                                             (N_NODES / 16) * 16);
  logits_kernel<256, 4><<<NB_N, 256, 0, stream>>>(h, a1s, a1d, als, ald);
  gat_aggregate<256, 4, 8><<<NB_N, 256, 0, stream>>>(h, als, ald, row_off, edge_srcs,
                                                     b1, bufA);
  ln_relu_kernel<<<NB_N, 256, 0, stream>>>(bufA, ln1g, ln1b);

  // --- Layer 2: [N,256]@[256,256] -> GAT -> LN+ReLU ---
  gemm_wmma_f32<<<(N_NODES / 16) * 16 / 8, 256, 0, stream>>>(bufA, W2, h, 256, 256, 16,
                                                             (N_NODES / 16) * 16);
  logits_kernel<256, 4><<<NB_N, 256, 0, stream>>>(h, a2s, a2d, als, ald);
  gat_aggregate<256, 4, 8><<<NB_N, 256, 0, stream>>>(h, als, ald, row_off, edge_srcs,
                                                     b2, bufB);
  ln_relu_kernel<<<NB_N, 256, 0, stream>>>(bufB, ln2g, ln2b);

  // --- Layer 3: [N,256]@[256,64] -> GAT (H=1, mean==identity) -> d_out ---
  const int t3 = (N_NODES / 16) * 4;  // 12500 tiles
  gemm_wmma_f32<<<(t3 + 7) / 8, 256, 0, stream>>>(bufB, W3, h, 256, 64, 4, t3);
  logits_kernel<64, 1><<<NB_N, 256, 0, stream>>>(h, a3s, a3d, als, ald);
  gat_aggregate<64, 1, 2><<<NB_N, 256, 0, stream>>>(h, als, ald, row_off, edge_srcs,
                                                    b3, out);
}